// PixelBarlowTwinsLoss_15264313770057
// MI455X (gfx1250) — compile-verified
//
#include <hip/hip_runtime.h>

typedef __attribute__((ext_vector_type(8))) float v8f;
typedef __attribute__((ext_vector_type(2))) float v2f;

#define D_CH   256
#define HW     16384
#define M_SAMP 4096
#define B_BATCH 16
#define CH     1024          // samples (K) per workgroup
#define KS     32            // K-tile staged in LDS per iteration
#define ROWP   34            // padded LDS row pitch (floats), even for b64 align
#define N_TOT  (B_BATCH * M_SAMP)     // 65536
#define G_ELEMS (D_CH * D_CH)         // 65536
#define WS_FLOATS (G_ELEMS + 4 * D_CH)

// ---------------- workspace zeroing (graph-replay safe) ----------------
__global__ void bt_zero_ws(float* ws) {
  int i = blockIdx.x * blockDim.x + threadIdx.x;
  if (i < WS_FLOATS) ws[i] = 0.0f;
}

// ---------------- fused gather + split-K fp32 WMMA GEMM ----------------
// ws layout: G[256*256], S0[256], SS0[256], S1[256], SS1[256]
__global__ __launch_bounds__(512) void bt_gemm(const float* __restrict__ z,
                                               const float* __restrict__ zp,
                                               const int* __restrict__ fidx,
                                               float* __restrict__ ws) {
  __shared__ int   pIdx[CH];
  __shared__ float At[D_CH * ROWP];   // Zm tile: [channel][k]   (34 KB)
  __shared__ float Bt[D_CH * ROWP];   // Zp tile: [channel][k]

  const int b     = blockIdx.x;
  const int chunk = blockIdx.y;
  const int tid   = threadIdx.x;

  for (int t = tid; t < CH; t += 512)
    pIdx[t] = fidx[b * M_SAMP + chunk * CH + t];

  // gather role: threads 0..255 gather z channels, 256..511 gather z'
  const int d   = tid & 255;
  const int isB = tid >> 8;
  const float* src = (isB ? zp : z) + ((size_t)(b * D_CH + d)) * HW;
  float* tile      = (isB ? Bt : At) + d * ROWP;

  // WMMA role: wave w owns output rows [16w, 16w+16)
  const int wave = tid >> 5;
  const int lane = tid & 31;
  const int half = lane >> 4;   // lanes 0-15 -> K{0,1}; 16-31 -> K{2,3}
  const int lr   = lane & 15;

  v8f acc[16];
#pragma unroll
  for (int jt = 0; jt < 16; ++jt) acc[jt] = (v8f)0.0f;

  float s = 0.0f, ss = 0.0f;
  __syncthreads();  // pIdx visible

  for (int ks = 0; ks < CH / KS; ++ks) {
    // ---- gather KS samples for this channel into LDS ----
#pragma unroll 4
    for (int kk = 0; kk < KS; ++kk) {
      const int   p = pIdx[ks * KS + kk];
      const float v = src[p];
      tile[kk] = v;
      s  += v;
      ss += v * v;
    }
    __syncthreads();

    // ---- 8 WMMA k-steps (K=4 each) x 16 column tiles ----
#pragma unroll
    for (int k4 = 0; k4 < KS / 4; ++k4) {
      const int kk = k4 * 4 + half * 2;
      const v2f a = *(const v2f*)&At[(wave * 16 + lr) * ROWP + kk];
#pragma unroll
      for (int jt = 0; jt < 16; ++jt) {
        const v2f bb = *(const v2f*)&Bt[(jt * 16 + lr) * ROWP + kk];
        acc[jt] = __builtin_amdgcn_wmma_f32_16x16x4_f32(
            false, a, false, bb, (short)0, acc[jt], false, false);
      }
    }
    __syncthreads();
  }

  // ---- column-sum partials (one thread per (matrix, channel)) ----
  atomicAdd(&ws[G_ELEMS + isB * 2 * D_CH + d], s);
  atomicAdd(&ws[G_ELEMS + isB * 2 * D_CH + D_CH + d], ss);

  // ---- split-K accumulate G ----
#pragma unroll
  for (int jt = 0; jt < 16; ++jt) {
#pragma unroll
    for (int r = 0; r < 8; ++r) {
      const int i = wave * 16 + half * 8 + r;  // C layout: M = r + 8*(lane>=16)
      const int j = jt * 16 + lr;              // N = lane & 15
      atomicAdd(&ws[i * D_CH + j], acc[jt][r]);
    }
  }
}

// ---------------- epilogue: normalize-on-the-fly + loss reduce ----------------
__global__ __launch_bounds__(256) void bt_loss(const float* __restrict__ ws,
                                               float* __restrict__ out) {
  __shared__ float muB[D_CH];
  __shared__ float invB[D_CH];
  __shared__ float red[D_CH];
  const int   i = threadIdx.x;
  const float N = (float)N_TOT;

  const float S0 = ws[G_ELEMS + i];
  const float Q0 = ws[G_ELEMS + D_CH + i];
  const float S1 = ws[G_ELEMS + 2 * D_CH + i];
  const float Q1 = ws[G_ELEMS + 3 * D_CH + i];

  const float mu  = S0 / N;
  float var0 = (Q0 - S0 * S0 / N) / (N - 1.0f);     // ddof=1
  float sd0  = fmaxf(sqrtf(fmaxf(var0, 0.0f)), 1e-6f);
  const float nu  = S1 / N;
  float var1 = (Q1 - S1 * S1 / N) / (N - 1.0f);
  float sd1  = fmaxf(sqrtf(fmaxf(var1, 0.0f)), 1e-6f);

  muB[i]  = nu;
  invB[i] = 1.0f / sd1;
  __syncthreads();

  const float invA = 1.0f / (N * sd0);
  float accd = 0.0f;
  for (int j = 0; j < D_CH; ++j) {
    const float C = (ws[i * D_CH + j] - N * mu * muB[j]) * invA * invB[j];
    if (j == i) { const float t = 1.0f - C; accd += t * t; }
    else        { accd += 0.005f * C * C; }
  }
  red[i] = accd;
  __syncthreads();
  for (int sft = 128; sft > 0; sft >>= 1) {
    if (i < sft) red[i] += red[i + sft];
    __syncthreads();
  }
  if (i == 0) out[0] = red[0];
}

extern "C" void kernel_launch(void* const* d_in, const int* in_sizes, int n_in,
                              void* d_out, int out_size, void* d_ws, size_t ws_size,
                              hipStream_t stream) {
  const float* z    = (const float*)d_in[0];
  const float* zp   = (const float*)d_in[1];
  const int*   fidx = (const int*)d_in[2];
  float* ws  = (float*)d_ws;   // needs WS_FLOATS*4 = ~260 KB
  float* out = (float*)d_out;

  bt_zero_ws<<<(WS_FLOATS + 255) / 256, 256, 0, stream>>>(ws);
  bt_gemm<<<dim3(B_BATCH, M_SAMP / CH), 512, 0, stream>>>(z, zp, fidx, ws);
  bt_loss<<<1, 256, 0, stream>>>(ws, out);
}